// GCN_64063732187750
// MI455X (gfx1250) — compile-verified
//
#include <hip/hip_runtime.h>
#include <hip/hip_bf16.h>
#include <math.h>

// ---------------------------------------------------------------------------
// GCN (3 graphs x 2 layers, N=100000, D=64, E=3.2M) for MI455X / gfx1250.
// GEMMs via V_WMMA_F32_16X16X4_F32 (fp32, wave32); scatter via f32 L2 atomics.
// Workspace layout (floats):
//   xm   [N*384]  interleaved [g1 g2 c1 c2 f1 f2] per row  (153.6 MB, ~fits L2)
//   h1   [N*64]   x@W1 (shared by all three layer-1 convs)
//   h2   [N*64]   per-graph layer-2 GEMM output
//   deg  [N], dinv [N], sums[8], att[8]
// ---------------------------------------------------------------------------

#define N_NODES 100000
#define N_EDGES 3200000
#define DIM     64
#define XM_COLS 384
#define DN      6400000LL  // DIM * N_NODES

typedef __attribute__((ext_vector_type(2))) float v2f;
typedef __attribute__((ext_vector_type(8))) float v8f;

__global__ void zero_f32(float* __restrict__ p, long long n) {
  long long i = (long long)blockIdx.x * blockDim.x + threadIdx.x;
  if (i < n) p[i] = 0.0f;
}

// deg[col] += ew  (self-loop weight 1 folded into dinv_kernel)
__global__ void deg_kernel(const int* __restrict__ col, const float* __restrict__ ew,
                           float* __restrict__ deg, int E) {
  int e = blockIdx.x * blockDim.x + threadIdx.x;
  if (e < E) atomicAdd(&deg[col[e]], ew[e]);
}

__global__ void dinv_kernel(const float* __restrict__ deg, float* __restrict__ dinv, int n) {
  int i = blockIdx.x * blockDim.x + threadIdx.x;
  if (i < n) dinv[i] = rsqrtf(deg[i] + 1.0f);  // deg+1 >= 1 always (self loop)
}

// C[nrows x 64] = A[nrows x 64 (lda)] @ W[64 x 64], fp32 WMMA 16x16x4.
// One wave computes a 16x64 output tile (4 accumulators of 16x16).
// W is staged in LDS pre-swizzled as k-pairs: sW2[p][n] = {W[2p][n], W[2p+1][n]}
// so each B operand is one aligned ds_load_b64 straight into a VGPR pair.
__global__ void __launch_bounds__(256)
gemm64_wmma(const float* __restrict__ A, int lda,
            const float* __restrict__ W,
            float* __restrict__ C, int ldc, int nrows) {
  __shared__ v2f sW2[(DIM / 2) * DIM];  // 32 k-pairs x 64 cols, 16 KB
  {
    float* sWf = (float*)sW2;
    for (int i = threadIdx.x; i < DIM * DIM; i += blockDim.x) {
      int kidx = i >> 6;            // source row k
      int n    = i & 63;            // source col
      int p    = kidx >> 1;         // k-pair index
      int half = kidx & 1;          // position within pair
      sWf[(p * DIM + n) * 2 + half] = W[i];
    }
  }
  __syncthreads();

  const int wave = threadIdx.x >> 5;
  const int lane = threadIdx.x & 31;
  const int row0 = (blockIdx.x * 8 + wave) * 16;
  const int m    = lane & 15;
  const int kh   = (lane >> 4) << 1;  // 0 (lanes 0-15) or 2 (lanes 16-31)

  int r = row0 + m;
  if (r > nrows - 1) r = nrows - 1;   // clamp loads, keep EXEC all-ones for WMMA
  const float* arow = A + (size_t)r * lda;

  v8f acc0 = {}, acc1 = {}, acc2 = {}, acc3 = {};
#pragma unroll
  for (int k = 0; k < DIM; k += 4) {
    // A 16x4 fp32 layout: v0 = {K=0 | K=2}, v1 = {K=1 | K=3} across half-waves
    // (k + kh) is even -> 8-byte aligned -> single global_load_b64
    v2f a = *(const v2f*)&arow[k + kh];
    const v2f* wp = &sW2[((k + kh) >> 1) * DIM + m];
    v2f b0 = wp[0];    // {W[k+kh][t*16+m], W[k+kh+1][t*16+m]} in one ds_load_b64
    v2f b1 = wp[16];
    v2f b2 = wp[32];
    v2f b3 = wp[48];
    acc0 = __builtin_amdgcn_wmma_f32_16x16x4_f32(false, a, false, b0, (short)0, acc0, false, false);
    acc1 = __builtin_amdgcn_wmma_f32_16x16x4_f32(false, a, false, b1, (short)0, acc1, false, false);
    acc2 = __builtin_amdgcn_wmma_f32_16x16x4_f32(false, a, false, b2, (short)0, acc2, false, false);
    acc3 = __builtin_amdgcn_wmma_f32_16x16x4_f32(false, a, false, b3, (short)0, acc3, false, false);
  }

  // C/D layout: VGPR v -> M = v + (lane<16 ? 0 : 8), N = lane&15
  const int srow0 = row0 + ((lane >> 4) << 3);
  const int nc    = lane & 15;
#pragma unroll
  for (int v = 0; v < 8; ++v) {
    int rr = srow0 + v;
    if (rr < nrows) {
      float* crow = C + (size_t)rr * ldc + nc;
      crow[0]  = acc0[v];
      crow[16] = acc1[v];
      crow[32] = acc2[v];
      crow[48] = acc3[v];
    }
  }
}

// 64 consecutive threads per edge (one feature dim each): coalesced gather of
// h[row,:], coalesced f32 atomics into xm[col, colbase:colbase+64]. All 32
// lanes of a wave share one edge -> the scalar loads broadcast-coalesce.
__global__ void scatter_edges(const int* __restrict__ row, const int* __restrict__ col,
                              const float* __restrict__ ew, const float* __restrict__ dinv,
                              const float* __restrict__ h, float* __restrict__ xm,
                              int colbase, int E) {
  long long gid = (long long)blockIdx.x * blockDim.x + threadIdx.x;
  long long e = gid >> 6;
  if (e >= E) return;
  int d = (int)(gid & 63);
  int r = row[e];
  int c = col[e];
  float nrm = dinv[r] * ew[e] * dinv[c];
  atomicAdd(&xm[(size_t)c * XM_COLS + colbase + d], nrm * h[(size_t)r * DIM + d]);
}

// out = relu(acc + dinv^2 * h + bias)   (dinv^2 = self-loop norm)
__global__ void finalize_layer(float* __restrict__ xm, const float* __restrict__ dinv,
                               const float* __restrict__ h, const float* __restrict__ bias,
                               int colbase) {
  long long i = (long long)blockIdx.x * blockDim.x + threadIdx.x;
  if (i >= DN) return;
  int n = (int)(i >> 6);
  int d = (int)(i & 63);
  float di = dinv[n];
  size_t idx = (size_t)n * XM_COLS + colbase + d;
  float v = xm[idx] + di * di * h[(size_t)n * DIM + d] + bias[d];
  xm[idx] = v > 0.0f ? v : 0.0f;
}

// Faithful channel means of XM.reshape(1,6,D,N): contiguous flat chunks of DN.
__global__ void channel_sums(const float* __restrict__ xm, float* __restrict__ sums) {
  float part[6] = {0.f, 0.f, 0.f, 0.f, 0.f, 0.f};
  const long long total = 6 * DN;
  const long long stride = (long long)gridDim.x * blockDim.x;
  for (long long i = (long long)blockIdx.x * blockDim.x + threadIdx.x; i < total; i += stride) {
    part[(int)(i / DN)] += xm[i];
  }
  __shared__ float red[256];
#pragma unroll
  for (int c = 0; c < 6; ++c) {
    red[threadIdx.x] = part[c];
    __syncthreads();
    for (int s = blockDim.x >> 1; s > 0; s >>= 1) {
      if ((int)threadIdx.x < s) red[threadIdx.x] += red[threadIdx.x + s];
      __syncthreads();
    }
    if (threadIdx.x == 0) atomicAdd(&sums[c], red[0]);
    __syncthreads();
  }
}

// SE attention MLP: 6 -> 30 (relu) -> 6 (sigmoid). Trivial size, one lane.
__global__ void att_kernel(const float* __restrict__ sums,
                           const float* __restrict__ fc1_w, const float* __restrict__ fc1_b,
                           const float* __restrict__ fc2_w, const float* __restrict__ fc2_b,
                           float* __restrict__ att) {
  if (threadIdx.x != 0 || blockIdx.x != 0) return;
  float m[6], h30[30];
  for (int c = 0; c < 6; ++c) m[c] = sums[c] / (float)DN;
  for (int j = 0; j < 30; ++j) {
    float s = fc1_b[j];
    for (int i = 0; i < 6; ++i) s += m[i] * fc1_w[j * 6 + i];
    h30[j] = s > 0.f ? s : 0.f;
  }
  for (int j = 0; j < 6; ++j) {
    float s = fc2_b[j];
    for (int i = 0; i < 30; ++i) s += h30[i] * fc2_w[j * 30 + i];
    att[j] = 1.0f / (1.0f + __expf(-s));
  }
}

// out[n,d] = conv_b + sum_c conv_w[c] * relu(att[c] * XMflat[c*DN + d*N + n])
// (exactly the reference's reshape(1,6,D,N) view semantics)
__global__ void combine_out(const float* __restrict__ xm, const float* __restrict__ att,
                            const float* __restrict__ cw, const float* __restrict__ cb,
                            float* __restrict__ out) {
  long long i = (long long)blockIdx.x * blockDim.x + threadIdx.x;
  if (i >= DN) return;
  int d = (int)(i / N_NODES);
  long long n = i % N_NODES;
  float s = cb[0];
#pragma unroll
  for (int c = 0; c < 6; ++c) {
    float v = att[c] * xm[(long long)c * DN + (long long)d * N_NODES + n];
    s += cw[c] * (v > 0.0f ? v : 0.0f);
  }
  out[n * DIM + d] = s;
}

extern "C" void kernel_launch(void* const* d_in, const int* in_sizes, int n_in,
                              void* d_out, int out_size, void* d_ws, size_t ws_size,
                              hipStream_t stream) {
  (void)in_sizes; (void)n_in; (void)out_size; (void)ws_size;
  const float* x     = (const float*)d_in[0];
  const float* ew[3] = {(const float*)d_in[1], (const float*)d_in[2], (const float*)d_in[3]};
  const int*   eg[3] = {(const int*)d_in[4], (const int*)d_in[5], (const int*)d_in[6]};
  const float* W1 = (const float*)d_in[7];
  const float* b1 = (const float*)d_in[8];
  const float* W2 = (const float*)d_in[9];
  const float* b2 = (const float*)d_in[10];
  const float* fc1_w = (const float*)d_in[11];
  const float* fc1_b = (const float*)d_in[12];
  const float* fc2_w = (const float*)d_in[13];
  const float* fc2_b = (const float*)d_in[14];
  const float* conv_w = (const float*)d_in[15];
  const float* conv_b = (const float*)d_in[16];
  float* out = (float*)d_out;

  float* ws   = (float*)d_ws;
  float* xm   = ws;                                // N*384
  float* h1   = xm + (size_t)N_NODES * XM_COLS;    // N*64
  float* h2   = h1 + (size_t)N_NODES * DIM;        // N*64
  float* deg  = h2 + (size_t)N_NODES * DIM;        // N
  float* dinv = deg + N_NODES;                     // N
  float* sums = dinv + N_NODES;                    // 8
  float* att  = sums + 8;                          // 8

  const int B = 256;
  const long long xmN = (long long)N_NODES * XM_COLS;
  zero_f32<<<(int)((xmN + B - 1) / B), B, 0, stream>>>(xm, xmN);
  zero_f32<<<1, 32, 0, stream>>>(sums, 8);

  const int gemm_blocks = (N_NODES + 127) / 128;  // 8 waves/block, 16 rows/wave
  gemm64_wmma<<<gemm_blocks, 256, 0, stream>>>(x, DIM, W1, h1, DIM, N_NODES);

  const long long scat_threads = (long long)N_EDGES * 64;
  const int scat_blocks = (int)((scat_threads + B - 1) / B);
  const int fin_blocks  = (int)((DN + B - 1) / B);

  for (int q = 0; q < 3; ++q) {
    const int* row = eg[q];
    const int* col = eg[q] + N_EDGES;
    const int cb1 = q * 128;
    const int cb2 = q * 128 + 64;

    zero_f32<<<(N_NODES + B - 1) / B, B, 0, stream>>>(deg, N_NODES);
    deg_kernel<<<(N_EDGES + B - 1) / B, B, 0, stream>>>(col, ew[q], deg, N_EDGES);
    dinv_kernel<<<(N_NODES + B - 1) / B, B, 0, stream>>>(deg, dinv, N_NODES);

    // layer 1 (h1 = x@W1 shared across graphs)
    scatter_edges<<<scat_blocks, B, 0, stream>>>(row, col, ew[q], dinv, h1, xm, cb1, N_EDGES);
    finalize_layer<<<fin_blocks, B, 0, stream>>>(xm, dinv, h1, b1, cb1);

    // layer 2: A = xm[:, cb1:cb1+64] strided (lda = 384)
    gemm64_wmma<<<gemm_blocks, 256, 0, stream>>>(xm + cb1, XM_COLS, W2, h2, DIM, N_NODES);
    scatter_edges<<<scat_blocks, B, 0, stream>>>(row, col, ew[q], dinv, h2, xm, cb2, N_EDGES);
    finalize_layer<<<fin_blocks, B, 0, stream>>>(xm, dinv, h2, b2, cb2);
  }

  channel_sums<<<1024, 256, 0, stream>>>(xm, sums);
  att_kernel<<<1, 32, 0, stream>>>(sums, fc1_w, fc1_b, fc2_w, fc2_b, att);
  combine_out<<<fin_blocks, B, 0, stream>>>(xm, att, conv_w, conv_b, out);
}